// RecyclingEmbedder_38774964748734
// MI455X (gfx1250) — compile-verified
//
#include <hip/hip_runtime.h>
#include <stdint.h>

#define NRES   1024
#define CM     256
#define CZ     128
#define NBINS  15
#define LN_EPS 1e-5f
#define NWAVES 8
#define NBUF   4      // async ring depth (prefetch distance 3)

typedef float v4f __attribute__((ext_vector_type(4)));
// Match the builtin's parameter type exactly: vector_size(16) int, AS1 / AS3.
typedef int vs4i __attribute__((vector_size(16)));
typedef __attribute__((address_space(1))) vs4i g_vs4i;   // global
typedef __attribute__((address_space(3))) vs4i l_vs4i;   // LDS

#ifndef __has_builtin
#define __has_builtin(x) 0
#endif

#if defined(__HIP_DEVICE_COMPILE__) && __has_builtin(__builtin_amdgcn_global_load_async_to_lds_b128)
#define ASYNC_BUILTIN 1
#else
#define ASYNC_BUILTIN 0
#endif

#if defined(__HIP_DEVICE_COMPILE__) && __has_builtin(__builtin_amdgcn_s_wait_asynccnt)
#define WAITASYNC_BUILTIN 1
#else
#define WAITASYNC_BUILTIN 0
#endif

// One 16-byte lane-wise async copy global -> LDS (tracked by ASYNCcnt).
__device__ __forceinline__ void async_b128_to_lds(const float* g, float* l) {
#if ASYNC_BUILTIN
  __builtin_amdgcn_global_load_async_to_lds_b128(
      (g_vs4i*)(uintptr_t)g,                 // global AS1 (generic value == AS1 value)
      (l_vs4i*)(uint32_t)(uintptr_t)l,       // LDS offset = addr[31:0] (ISA 10.2)
      0, 0);
#else
  unsigned long long ga = (unsigned long long)(uintptr_t)g;
  uint32_t la = (uint32_t)(uintptr_t)l;
  asm volatile("global_load_async_to_lds_b128 %0, %1, off"
               :: "v"(la), "v"(ga) : "memory");
#endif
}

template <int NW>
__device__ __forceinline__ void wait_asynccnt_le() {
#if WAITASYNC_BUILTIN
  __builtin_amdgcn_s_wait_asynccnt(NW);
#else
  asm volatile("s_wait_asynccnt %0" :: "i"(NW) : "memory");
#endif
  asm volatile("" ::: "memory");  // keep LDS reads after the wait
}

__device__ __forceinline__ float wave_sum32(float v) {
#pragma unroll
  for (int m = 16; m > 0; m >>= 1) v += __shfl_xor(v, m, 32);
  return v;
}

// ---------------------------------------------------------------------------
// Pair kernel: block = row i (1024 blocks, 256 threads = 8 waves).
// new_pair[i,j,c] = table[idx(i,j)][c] + LN(pair[i,j,:])[c]
// ---------------------------------------------------------------------------
__global__ __launch_bounds__(256) void recycle_pair_kernel(
    const float* __restrict__ pair, const float* __restrict__ cb,
    const float* __restrict__ W,    const float* __restrict__ bvec,
    const float* __restrict__ lnw,  const float* __restrict__ lnb,
    const float* __restrict__ bins, float* __restrict__ out)
{
  __shared__ __align__(16) float tbl[NBINS * CZ];        // W.T[k][c] + b[c]
  __shared__ __align__(16) float cbs[NRES * 3];
  __shared__ float binss[NBINS];
  __shared__ __align__(16) float stage[NWAVES][NBUF][CZ];  // per-wave ring

  const int tid = threadIdx.x;
  const int i   = blockIdx.x;

  // Stage embedding table: thread c owns contiguous W row (loads fully unrolled).
  if (tid < CZ) {
    const float bc = bvec[tid];
    float wrow[NBINS];
#pragma unroll
    for (int k = 0; k < NBINS; ++k) wrow[k] = W[tid * NBINS + k];
#pragma unroll
    for (int k = 0; k < NBINS; ++k) tbl[k * CZ + tid] = wrow[k] + bc;
  }
  // Stage CB coords as vectorized copy (1024*3 floats = 768 x b128).
  {
    const v4f* cb4 = (const v4f*)cb;
    v4f* cbs4 = (v4f*)cbs;
    for (int t = tid; t < (NRES * 3) / 4; t += 256) cbs4[t] = cb4[t];
  }
  if (tid < NBINS) binss[tid] = bins[tid];
  __syncthreads();

  const int lane = tid & 31;
  const int wid  = tid >> 5;
  const int c0   = lane << 2;

  const v4f wv = *(const v4f*)(lnw + c0);
  const v4f bv = *(const v4f*)(lnb + c0);
  const float cix = cbs[i * 3 + 0];
  const float ciy = cbs[i * 3 + 1];
  const float ciz = cbs[i * 3 + 2];

  const float* rowbase = pair + (size_t)i * NRES * CZ;
  float*       outbase = out  + (size_t)i * NRES * CZ;

  // Prime pipeline with prefetch distance 3: wave owns j = wid + 8*t.
#pragma unroll
  for (int p = 0; p < NBUF - 1; ++p)
    async_b128_to_lds(rowbase + (size_t)(wid + 8 * p) * CZ + c0,
                      &stage[wid][p][c0]);

  const int TRIPS = NRES / NWAVES;  // 128 iterations, uniform per wave
  for (int t = 0; t < TRIPS; ++t) {
    const int j   = wid + (t << 3);
    const int rem = (TRIPS - 1) - t;
    // Issue prefetch for t+3, then wait so the oldest (current) row is done.
    // ASYNCcnt retires in order; tail must tighten the wait level.
    if (rem >= 3) {
      async_b128_to_lds(rowbase + (size_t)(j + 24) * CZ + c0,
                        &stage[wid][(t + 3) & (NBUF - 1)][c0]);
      wait_asynccnt_le<3>();
    } else if (rem == 2) {
      wait_asynccnt_le<2>();
    } else if (rem == 1) {
      wait_asynccnt_le<1>();
    } else {
      wait_asynccnt_le<0>();
    }

    const v4f x = *(const v4f*)(&stage[wid][t & (NBUF - 1)][c0]);

    // distance -> nearest bin (first-min tie break, like jnp.argmin)
    const float dx = cix - cbs[j * 3 + 0];
    const float dy = ciy - cbs[j * 3 + 1];
    const float dz = ciz - cbs[j * 3 + 2];
    const float dist = sqrtf(dx * dx + dy * dy + dz * dz);
    int best = 0;
    float bd = fabsf(dist - binss[0]);
#pragma unroll
    for (int k = 1; k < NBINS; ++k) {
      const float d = fabsf(dist - binss[k]);
      if (d < bd) { bd = d; best = k; }
    }

    float s = x.x + x.y + x.z + x.w;
    float q = x.x * x.x + x.y * x.y + x.z * x.z + x.w * x.w;
    s = wave_sum32(s);
    q = wave_sum32(q);
    const float mu  = s * (1.0f / CZ);
    const float var = q * (1.0f / CZ) - mu * mu;
    const float rs  = rsqrtf(var + LN_EPS);

    const v4f tv = *(const v4f*)(&tbl[best * CZ + c0]);
    v4f y = (x - mu) * rs;
    y = y * wv + bv + tv;

    __builtin_nontemporal_store(y, (v4f*)(outbase + (size_t)j * CZ + c0));
  }
}

// ---------------------------------------------------------------------------
// MSA kernel: one wave per 256-wide row (128 blocks x 8 waves).
// ---------------------------------------------------------------------------
__global__ __launch_bounds__(256) void recycle_msa_kernel(
    const float* __restrict__ x, const float* __restrict__ w,
    const float* __restrict__ b, float* __restrict__ out)
{
  const int row  = (blockIdx.x << 3) + (threadIdx.x >> 5);
  const int lane = threadIdx.x & 31;
  const int c0   = lane << 2;

  const float* xr = x + (size_t)row * CM;
  const v4f a = *(const v4f*)(xr + c0);
  const v4f d = *(const v4f*)(xr + c0 + 128);

  float s = a.x + a.y + a.z + a.w + d.x + d.y + d.z + d.w;
  float q = a.x * a.x + a.y * a.y + a.z * a.z + a.w * a.w +
            d.x * d.x + d.y * d.y + d.z * d.z + d.w * d.w;
  s = wave_sum32(s);
  q = wave_sum32(q);
  const float mu  = s * (1.0f / CM);
  const float var = q * (1.0f / CM) - mu * mu;
  const float rs  = rsqrtf(var + LN_EPS);

  const v4f wa = *(const v4f*)(w + c0);
  const v4f wd = *(const v4f*)(w + c0 + 128);
  const v4f ba = *(const v4f*)(b + c0);
  const v4f bdv = *(const v4f*)(b + c0 + 128);

  v4f ya = (a - mu) * rs; ya = ya * wa + ba;
  v4f yd = (d - mu) * rs; yd = yd * wd + bdv;

  float* orow = out + (size_t)row * CM;
  __builtin_nontemporal_store(ya, (v4f*)(orow + c0));
  __builtin_nontemporal_store(yd, (v4f*)(orow + c0 + 128));
}

// ---------------------------------------------------------------------------
extern "C" void kernel_launch(void* const* d_in, const int* in_sizes, int n_in,
                              void* d_out, int out_size, void* d_ws, size_t ws_size,
                              hipStream_t stream) {
  (void)in_sizes; (void)n_in; (void)out_size; (void)d_ws; (void)ws_size;

  const float* msa   = (const float*)d_in[0];  // (1024,256)
  const float* pair  = (const float*)d_in[1];  // (1024,1024,128)
  const float* cb    = (const float*)d_in[2];  // (1024,3)
  const float* W     = (const float*)d_in[3];  // (128,15)
  const float* bvec  = (const float*)d_in[4];  // (128,)
  const float* lnpw  = (const float*)d_in[5];
  const float* lnpb  = (const float*)d_in[6];
  const float* lnmw  = (const float*)d_in[7];
  const float* lnmb  = (const float*)d_in[8];
  const float* bins  = (const float*)d_in[9];

  float* out_msa  = (float*)d_out;                      // first output
  float* out_pair = (float*)d_out + (size_t)NRES * CM;  // second output

  recycle_msa_kernel<<<NRES / 8, 256, 0, stream>>>(msa, lnmw, lnmb, out_msa);
  recycle_pair_kernel<<<NRES, 256, 0, stream>>>(pair, cb, W, bvec, lnpw, lnpb,
                                                bins, out_pair);
}